// GraphEncoderNetwork_45337674777313
// MI455X (gfx1250) — compile-verified
//
#include <hip/hip_runtime.h>
#include <hip/hip_bf16.h>

// ---------------------------------------------------------------------------
// CDNA5 (gfx1250) GNN pipeline. All matmuls via V_WMMA_F32_16X16X4_F32.
// One wave32 owns a 16-row tile; fragments per CDNA5 ISA layouts:
//   A 16x4 f32 : lane L holds row M=L&15, K = {0,1}+2*(L>>4)      (v2f)
//   B 4x16 f32 : lane L holds col N=L&15, K = {0,1}+2*(L>>4)      (v2f)
//   C/D 16x16  : lane L holds col N=L&15, rows M = r + 8*(L>>4)   (v8f)
// Activations are zero-padded to K1=roundup4(KIN) in LDS and weights are
// pre-swizzled into B-fragment order so every inner-loop LDS access is an
// unconditional, conflict-free ds_load_b64 (no EXEC-predicated loads).
// ---------------------------------------------------------------------------

typedef __attribute__((ext_vector_type(2))) float v2f;
typedef __attribute__((ext_vector_type(8))) float v8f;

__device__ __forceinline__ v8f wmma4(v2f a, v2f b, v8f c) {
  // 8 args: (neg_a, A, neg_b, B, c_mod, C, reuse_a, reuse_b)
  return __builtin_amdgcn_wmma_f32_16x16x4_f32(false, a, false, b, (short)0, c,
                                               false, false);
}

struct Mlp  { const float *W1, *b1, *W2, *b2, *W3, *b3; };   // global, row-major
struct MlpS { const float *W1f, *b1, *W2f, *b2, *W3f, *b3; };// smem, frag layout

#define SX 24                                  // xs row stride (max K1 = 24)
#define SCR_PER_WAVE (384 + 512 + 256 + 256)   // xs + h1 + h2 + out

__device__ __forceinline__ v2f ld2(const float* p) { return *(const v2f*)p; }

__device__ __forceinline__ void ld_smem(float* dst, const float* src, int n,
                                        int tid, int nthr) {
  for (int i = tid; i < n; i += nthr) dst[i] = src[i];
}

// stage W [KIN x NOUT] row-major -> B-fragment layout, zero-padded to K1 rows.
// linear layout: (((s * NT + t) * 32 + lane) * 2 + j), where the pair {j=0,1}
// holds W[4s + 2*(lane>>4) + j][16t + (lane&15)].
template <int KIN, int NOUT>
__device__ __forceinline__ void stage_wfrag(float* dst,
                                            const float* __restrict__ W,
                                            int tid, int nthr) {
  constexpr int K1 = (KIN + 3) & ~3;
  constexpr int NT = NOUT / 16;
  constexpr int TOT = (K1 / 4) * NT * 64;
  for (int i = tid; i < TOT; i += nthr) {
    const int j  = i & 1;
    const int L  = (i >> 1) & 31;
    const int st = i >> 6;
    const int t  = st % NT, s = st / NT;
    const int krow = 4 * s + ((L >> 4) << 1) + j;
    const int c    = 16 * t + (L & 15);
    dst[i] = (krow < KIN) ? W[krow * NOUT + c] : 0.f;
  }
}

template <int KIN>
__device__ __forceinline__ MlpS stage_mlp(float* s, const Mlp& g, int tid,
                                          int nthr) {
  constexpr int K1 = (KIN + 3) & ~3;
  MlpS r;
  r.W1f = s; stage_wfrag<KIN, 32>(s, g.W1, tid, nthr); s += (K1 / 4) * 2 * 64;
  r.b1  = s; ld_smem(s, g.b1, 32, tid, nthr);          s += 32;
  r.W2f = s; stage_wfrag<32, 16>(s, g.W2, tid, nthr);  s += 8 * 64;
  r.b2  = s; ld_smem(s, g.b2, 16, tid, nthr);          s += 16;
  r.W3f = s; stage_wfrag<16, 16>(s, g.W3, tid, nthr);  s += 4 * 64;
  r.b3  = s; ld_smem(s, g.b3, 16, tid, nthr);
  return r;
}

// 3-layer MLP on one 16-row tile: [16,KIN] -> relu32 -> relu16 -> 16
// xs must be zero-padded to K1 columns.
template <int KIN>
__device__ __forceinline__ void mlp3_tile(const float* __restrict__ xs,
                                          const MlpS& m,
                                          float* __restrict__ h1,
                                          float* __restrict__ h2,
                                          float* __restrict__ out, int lane) {
  const int col = lane & 15;
  const int hi2 = (lane >> 4) << 1;   // 0 or 2
  constexpr int K1 = (KIN + 3) & ~3;
  // ------ layer 1: X[16,K1] @ W1[K1,32] + b1, ReLU ------
  v8f acc0, acc1;
  {
    const float bA = m.b1[col], bB = m.b1[col + 16];
#pragma unroll
    for (int r = 0; r < 8; ++r) { acc0[r] = bA; acc1[r] = bB; }
  }
#pragma unroll
  for (int s = 0; s < K1 / 4; ++s) {
    const v2f a  = ld2(xs + col * SX + 4 * s + hi2);
    const v2f b0 = ld2(m.W1f + (((2 * s + 0) * 32 + lane) << 1));
    const v2f b1 = ld2(m.W1f + (((2 * s + 1) * 32 + lane) << 1));
    acc0 = wmma4(a, b0, acc0);
    acc1 = wmma4(a, b1, acc1);
  }
#pragma unroll
  for (int r = 0; r < 8; ++r) {
    const int mrow = r + (hi2 << 2);
    h1[mrow * 32 + col]      = fmaxf(acc0[r], 0.f);
    h1[mrow * 32 + col + 16] = fmaxf(acc1[r], 0.f);
  }
  // ------ layer 2: H1[16,32] @ W2[32,16] + b2, ReLU ------
  v8f acc;
  {
    const float b2v = m.b2[col];
#pragma unroll
    for (int r = 0; r < 8; ++r) acc[r] = b2v;
  }
#pragma unroll
  for (int s = 0; s < 8; ++s) {
    const v2f a = ld2(h1 + col * 32 + 4 * s + hi2);
    const v2f b = ld2(m.W2f + ((s * 32 + lane) << 1));
    acc = wmma4(a, b, acc);
  }
#pragma unroll
  for (int r = 0; r < 8; ++r)
    h2[(r + (hi2 << 2)) * 16 + col] = fmaxf(acc[r], 0.f);
  // ------ layer 3: H2[16,16] @ W3[16,16] + b3 ------
  v8f acc3;
  {
    const float b3v = m.b3[col];
#pragma unroll
    for (int r = 0; r < 8; ++r) acc3[r] = b3v;
  }
#pragma unroll
  for (int s = 0; s < 4; ++s) {
    const v2f a = ld2(h2 + col * 16 + 4 * s + hi2);
    const v2f b = ld2(m.W3f + ((s * 32 + lane) << 1));
    acc3 = wmma4(a, b, acc3);
  }
#pragma unroll
  for (int r = 0; r < 8; ++r) out[(r + (hi2 << 2)) * 16 + col] = acc3[r];
}

// ---------------------------------------------------------------------------
__global__ void k_zero(float* __restrict__ p, int n) {
  int i = blockIdx.x * blockDim.x + threadIdx.x;
  if (i < n) p[i] = 0.f;
}

// prep MLP: x[N,6] -> h[N,16]
__global__ __launch_bounds__(256) void k_prep(const float* __restrict__ x,
                                              float* __restrict__ h, Mlp mp,
                                              int nTiles, int N) {
  __shared__ float sw[2 * 2 * 64 + 32 + 512 + 16 + 256 + 16];
  __shared__ float scr[8][SCR_PER_WAVE];
  const int tid = threadIdx.x;
  MlpS sm = stage_mlp<6>(sw, mp, tid, 256);
  __syncthreads();
  const int wave = tid >> 5, lane = tid & 31;
  const int tile = blockIdx.x * 8 + wave;
  if (tile >= nTiles) return;
  float* xs = scr[wave];
  float* h1 = xs + 384; float* h2 = h1 + 512; float* out = h2 + 256;
  for (int i = lane; i < 16 * 8; i += 32) {   // K1(6)=8, zero-pad cols 6,7
    int r = i >> 3, c = i & 7, node = tile * 16 + r;
    xs[r * SX + c] = (c < 6 && node < N) ? x[(size_t)node * 6 + c] : 0.f;
  }
  mlp3_tile<6>(xs, sm, h1, h2, out, lane);
  for (int i = lane; i < 256; i += 32) {
    int r = i >> 4, c = i & 15, node = tile * 16 + r;
    if (node < N) h[(size_t)node * 16 + c] = out[r * 16 + c];
  }
}

// edge scatter: x_agg[dst] += h[src]  (8M edges; h & x_agg live in 192MB L2)
__global__ __launch_bounds__(256) void k_edge(const int* __restrict__ src,
                                              const int* __restrict__ dst,
                                              const float* __restrict__ h,
                                              float* __restrict__ xagg, int E) {
  int e = blockIdx.x * blockDim.x + threadIdx.x;
  if (e >= E) return;
  const int s = src[e], d = dst[e];
  const float4* hs = (const float4*)(h + (size_t)s * 16);
  float* o = xagg + (size_t)d * 16;
#pragma unroll
  for (int q = 0; q < 4; ++q) {
    float4 v = hs[q];
    atomicAdd(o + q * 4 + 0, v.x);
    atomicAdd(o + q * 4 + 1, v.y);
    atomicAdd(o + q * 4 + 2, v.z);
    atomicAdd(o + q * 4 + 3, v.w);
  }
}

// fused: a = mlp_agg(x_agg); node_emb = mlp_proc(x||a);
//        merged = mlp_node(x||node_emb); dag_emb[batch] += merged
__global__ __launch_bounds__(256) void k_node(
    const float* __restrict__ x, const float* __restrict__ xagg,
    const int* __restrict__ batch, Mlp aggm, Mlp procm, Mlp nodem,
    float* __restrict__ node_out, float* __restrict__ dag_out, int nTiles,
    int N) {
  __shared__ float swA[4 * 2 * 64 + 32 + 512 + 16 + 256 + 16];
  __shared__ float swP[6 * 2 * 64 + 32 + 512 + 16 + 256 + 16];
  __shared__ float swN[6 * 2 * 64 + 32 + 512 + 16 + 256 + 16];
  __shared__ float scr[8][SCR_PER_WAVE];
  const int tid = threadIdx.x;
  MlpS sA = stage_mlp<16>(swA, aggm, tid, 256);
  MlpS sP = stage_mlp<22>(swP, procm, tid, 256);
  MlpS sN = stage_mlp<22>(swN, nodem, tid, 256);
  __syncthreads();
  const int wave = tid >> 5, lane = tid & 31;
  const int tile = blockIdx.x * 8 + wave;
  if (tile >= nTiles) return;
  float* xs = scr[wave];
  float* h1 = xs + 384; float* h2 = h1 + 512; float* out = h2 + 256;

  // stage 1: mlp_agg on x_agg tile (K1=16, no pad)
  for (int i = lane; i < 256; i += 32) {
    int r = i >> 4, c = i & 15, node = tile * 16 + r;
    xs[r * SX + c] = (node < N) ? xagg[(size_t)node * 16 + c] : 0.f;
  }
  mlp3_tile<16>(xs, sA, h1, h2, out, lane);   // out = a [16,16]

  // stage 2: mlp_proc on [x || a] (K1(22)=24, zero-pad cols 22,23)
  for (int i = lane; i < 16 * 24; i += 32) {
    int r = i / 24, c = i % 24, node = tile * 16 + r;
    float v = 0.f;
    if (c < 6)       v = (node < N) ? x[(size_t)node * 6 + c] : 0.f;
    else if (c < 22) v = out[r * 16 + (c - 6)];
    xs[r * SX + c] = v;
  }
  mlp3_tile<22>(xs, sP, h1, h2, out, lane);   // out = node_emb
  for (int i = lane; i < 256; i += 32) {
    int r = i >> 4, c = i & 15, node = tile * 16 + r;
    if (node < N) node_out[(size_t)node * 16 + c] = out[r * 16 + c];
  }

  // stage 3: mlp_node on [x || node_emb]
  for (int i = lane; i < 16 * 24; i += 32) {
    int r = i / 24, c = i % 24, node = tile * 16 + r;
    float v = 0.f;
    if (c < 6)       v = (node < N) ? x[(size_t)node * 6 + c] : 0.f;
    else if (c < 22) v = out[r * 16 + (c - 6)];
    xs[r * SX + c] = v;
  }
  mlp3_tile<22>(xs, sN, h1, h2, out, lane);   // out = merged

  // scatter into dag_emb; batch is sorted -> fast path when tile is uniform
  const int n0 = tile * 16, nLast = min(n0 + 15, N - 1);
  const int b0 = batch[n0], bL = batch[nLast];
  if (b0 == bL && n0 + 15 < N) {
    if (lane < 16) {
      float s = 0.f;
#pragma unroll
      for (int r = 0; r < 16; ++r) s += out[r * 16 + lane];
      atomicAdd(&dag_out[(size_t)b0 * 16 + lane], s);
    }
  } else {
    for (int i = lane; i < 256; i += 32) {
      int r = i >> 4, c = i & 15, node = n0 + r;
      if (node < N)
        atomicAdd(&dag_out[(size_t)batch[node] * 16 + c], out[r * 16 + c]);
    }
  }
}

// dag MLP + searchsorted scatter into z_sums
__global__ __launch_bounds__(256) void k_dag(const float* __restrict__ dag_emb,
                                             const int* __restrict__ indptr,
                                             Mlp dagm,
                                             float* __restrict__ zsums,
                                             int nTiles, int G, int O) {
  __shared__ float sw[4 * 2 * 64 + 32 + 512 + 16 + 256 + 16];
  __shared__ float scr[8][SCR_PER_WAVE];
  __shared__ int sInd[1024];
  const int tid = threadIdx.x;
  MlpS sm = stage_mlp<16>(sw, dagm, tid, 256);
  for (int i = tid; i <= O && i < 1024; i += 256) sInd[i] = indptr[i];
  __syncthreads();
  const int wave = tid >> 5, lane = tid & 31;
  const int tile = blockIdx.x * 8 + wave;
  if (tile >= nTiles) return;
  float* xs = scr[wave];
  float* h1 = xs + 384; float* h2 = h1 + 512; float* out = h2 + 256;
  for (int i = lane; i < 256; i += 32) {
    int r = i >> 4, c = i & 15, g = tile * 16 + r;
    xs[r * SX + c] = (g < G) ? dag_emb[(size_t)g * 16 + c] : 0.f;
  }
  mlp3_tile<16>(xs, sm, h1, h2, out, lane);
  for (int i = lane; i < 256; i += 32) {
    int r = i >> 4, c = i & 15, g = tile * 16 + r;
    if (g < G) {
      // seg = searchsorted(indptr, g, 'right') - 1 = max idx with indptr<=g
      int lo = 0, hiI = O;
      while (lo < hiI) {
        int mid = (lo + hiI + 1) >> 1;
        if (sInd[mid] <= g) lo = mid; else hiI = mid - 1;
      }
      atomicAdd(&zsums[(size_t)lo * 16 + c], out[r * 16 + c]);
    }
  }
}

__global__ void k_final(const float* __restrict__ zsums,
                        const int* __restrict__ indptr, float* __restrict__ z,
                        int O) {
  int i = blockIdx.x * blockDim.x + threadIdx.x;
  if (i < O * 16) {
    int o = i >> 4;
    int cnt = indptr[o + 1] - indptr[o];
    z[i] = zsums[i] / (float)max(cnt, 1);
  }
}

// ---------------------------------------------------------------------------
static Mlp mk_leaf(void* const* d_in, int base) {
  Mlp m;
  m.W1 = (const float*)d_in[base + 0];
  m.W2 = (const float*)d_in[base + 1];
  m.W3 = (const float*)d_in[base + 2];
  m.b1 = (const float*)d_in[base + 3];
  m.b2 = (const float*)d_in[base + 4];
  m.b3 = (const float*)d_in[base + 5];
  return m;
}
static Mlp mk_cat(const void* wp, const void* bp, int kin) {
  const float* W = (const float*)wp;
  const float* B = (const float*)bp;
  Mlp m;
  m.W1 = W;            m.W2 = W + kin * 32;   m.W3 = m.W2 + 32 * 16;
  m.b1 = B;            m.b2 = B + 32;         m.b3 = m.b2 + 16;
  return m;
}

extern "C" void kernel_launch(void* const* d_in, const int* in_sizes, int n_in,
                              void* d_out, int out_size, void* d_ws,
                              size_t ws_size, hipStream_t stream) {
  const float* x      = (const float*)d_in[0];
  const int*   ei     = (const int*)d_in[1];   // jax x64 disabled -> int32
  const int*   batch  = (const int*)d_in[2];
  const int*   indptr = (const int*)d_in[3];

  const int N = in_sizes[0] / 6;
  const int E = in_sizes[1] / 2;
  const int O = in_sizes[3] - 1;
  const int G = out_size / 16 - N - O;

  Mlp prep, proc, agg, node, dag;
  if (n_in >= 34) {  // pytree-leaf flattening: Ws then bs per MLP
    prep = mk_leaf(d_in, 4);
    proc = mk_leaf(d_in, 10);
    agg  = mk_leaf(d_in, 16);
    node = mk_leaf(d_in, 22);
    dag  = mk_leaf(d_in, 28);
  } else {           // list-concatenated flattening
    prep = mk_cat(d_in[4],  d_in[5],  6);
    proc = mk_cat(d_in[6],  d_in[7],  22);
    agg  = mk_cat(d_in[8],  d_in[9],  16);
    node = mk_cat(d_in[10], d_in[11], 22);
    dag  = mk_cat(d_in[12], d_in[13], 16);
  }

  const int* src = ei;
  const int* dst = ei + E;

  float* h      = (float*)d_ws;                 // [N,16]
  float* xagg   = h + (size_t)N * 16;           // [N,16]
  float* zsums  = xagg + (size_t)N * 16;        // [O,16]

  float* node_out = (float*)d_out;              // [N,16]
  float* dag_out  = node_out + (size_t)N * 16;  // [G,16]
  float* z        = dag_out + (size_t)G * 16;   // [O,16]

  const int nTilesN = (N + 15) / 16;
  const int nTilesG = (G + 15) / 16;

  // zero accumulators (every call; harness does not re-poison between replays)
  k_zero<<<(N * 16 + 255) / 256, 256, 0, stream>>>(xagg, N * 16);
  k_zero<<<(G * 16 + 255) / 256, 256, 0, stream>>>(dag_out, G * 16);
  k_zero<<<(O * 16 + 255) / 256, 256, 0, stream>>>(zsums, O * 16);

  k_prep<<<(nTilesN + 7) / 8, 256, 0, stream>>>(x, h, prep, nTilesN, N);
  k_edge<<<(E + 255) / 256, 256, 0, stream>>>(src, dst, h, xagg, E);
  k_node<<<(nTilesN + 7) / 8, 256, 0, stream>>>(x, xagg, batch, agg, proc,
                                                node, node_out, dag_out,
                                                nTilesN, N);
  k_dag<<<(nTilesG + 7) / 8, 256, 0, stream>>>(dag_out, indptr, dag, zsums,
                                               nTilesG, G, O);
  k_final<<<(O * 16 + 255) / 256, 256, 0, stream>>>(zsums, indptr, z, O);
}